// CoxPHLoss_23055384445702
// MI455X (gfx1250) — compile-verified
//
#include <hip/hip_runtime.h>

#define K_SEG 10000
#define THREADS 256
#define BLOCKS 512
#define TILE 2048   // elements per TDM tile (8 KB per array)

typedef __attribute__((ext_vector_type(2)))  float        v2f;
typedef __attribute__((ext_vector_type(8)))  float        v8f;
typedef __attribute__((ext_vector_type(4)))  unsigned int v4u;
typedef __attribute__((ext_vector_type(8)))  int          v8i;
typedef __attribute__((ext_vector_type(4)))  int          v4i;

// ---------------------------------------------------------------------------
// Kernel 0: zero the workspace accumulators (harness does not re-poison).
// ---------------------------------------------------------------------------
__global__ void zero_ws_kernel(float* __restrict__ p, int n) {
  int i = blockIdx.x * blockDim.x + threadIdx.x;
  if (i < n) p[i] = 0.0f;
}

// ---------------------------------------------------------------------------
// Kernel 1: per-duration segment sums of exp(log_h) and events.
// 80 KB dynamic LDS (two K_SEG float histograms) per workgroup -> fits the
// 320 KB CDNA5 WGP LDS; ds_add_f32 privatized atomics, one global flush per
// block. Single HBM-bound pass over 96 MB (~4.1 us at 23.3 TB/s).
// ---------------------------------------------------------------------------
__global__ void seg_hist_kernel(const float* __restrict__ log_h,
                                const int* __restrict__ durations,
                                const int* __restrict__ events,
                                float* __restrict__ g_exp,
                                float* __restrict__ g_ev,
                                int n) {
  extern __shared__ float smem[];
  float* s_exp = smem;
  float* s_ev  = smem + K_SEG;
  for (int k = threadIdx.x; k < K_SEG; k += blockDim.x) {
    s_exp[k] = 0.0f;
    s_ev[k]  = 0.0f;
  }
  __syncthreads();

  const int stride = gridDim.x * blockDim.x;
  for (int i = blockIdx.x * blockDim.x + threadIdx.x; i < n; i += stride) {
    __builtin_prefetch(log_h + i + 8 * stride, 0, 1);  // global_prefetch_b8
    const int d = durations[i];
    unsafeAtomicAdd(&s_exp[d], __expf(log_h[i]));
    if (events[i] != 0) unsafeAtomicAdd(&s_ev[d], 1.0f);
  }
  __syncthreads();

  for (int k = threadIdx.x; k < K_SEG; k += blockDim.x) {
    const float a = s_exp[k];
    const float b = s_ev[k];
    if (a != 0.0f) unsafeAtomicAdd(&g_exp[k], a);
    if (b != 0.0f) unsafeAtomicAdd(&g_ev[k], b);
  }
}

// ---------------------------------------------------------------------------
// Kernel 2: suffix cumsum over K_SEG bins -> risk; base = ev/risk (0 if
// risk<=0); also total event count. Single 1024-thread block.
// ---------------------------------------------------------------------------
__global__ void baseline_kernel(const float* __restrict__ g_exp,
                                const float* __restrict__ g_ev,
                                float* __restrict__ base,
                                float* __restrict__ tot_ev) {
  __shared__ float csum[1024];
  __shared__ float evs[1024];
  const int t  = threadIdx.x;
  const int CH = (K_SEG + 1023) / 1024;
  int k0 = t * CH;
  int k1 = k0 + CH;
  if (k1 > K_SEG) k1 = K_SEG;

  float s = 0.0f, ev = 0.0f;
  if (k0 < K_SEG) {
    for (int k = k0; k < k1; ++k) { s += g_exp[k]; ev += g_ev[k]; }
  }
  csum[t] = s;
  evs[t]  = ev;
  __syncthreads();

  if (t == 0) {
    float run = 0.0f, tot = 0.0f;
    for (int b = 1023; b >= 0; --b) {
      const float tmp = csum[b];
      csum[b] = run;            // exclusive suffix (sum of chunks after b)
      run += tmp;
      tot += evs[b];
    }
    tot_ev[0] = tot;
  }
  __syncthreads();

  if (k0 < K_SEG) {
    float run = csum[t];
    for (int k = k1 - 1; k >= k0; --k) {
      run += g_exp[k];          // risk[k] = sum over durations >= k
      base[k] = (run > 0.0f) ? (g_ev[k] / run) : 0.0f;
    }
  }
}

// ---------------------------------------------------------------------------
// TDM: issue a 1-row tensor_load_to_lds of `tile_elems` 4-byte elements from
// gptr into LDS at lds_byte_off. tensor_dim0 = valid_elems so out-of-tensor
// reads are hardware zero-filled (tail tiles). Descriptor per CDNA5 ISA ch.8:
//   group0: count=1 | lds_addr | global_addr[56:0] | type=2
//   group1: data_size=4B | tensor_dim0 | tensor_dim1=1 | tile_dim0 | tile_dim1=1
//   groups 2/3: zero (<=2D tensor)
// This toolchain uses the 6-arg builtin: (v4u, v8i, v4i, v4i, v8i, i32 cpol).
// Operands feed SGPR groups -> readfirstlane the wave-uniform scalars.
// ---------------------------------------------------------------------------
__device__ __forceinline__ void tdm_load_1d(const void* gptr,
                                            unsigned lds_byte_off,
                                            unsigned valid_elems,
                                            unsigned tile_elems) {
  const unsigned long long ga = (unsigned long long)gptr;
  const unsigned ga_lo = (unsigned)__builtin_amdgcn_readfirstlane((int)(unsigned)ga);
  const unsigned ga_hi = (unsigned)__builtin_amdgcn_readfirstlane((int)(unsigned)(ga >> 32));
  const unsigned lof   = (unsigned)__builtin_amdgcn_readfirstlane((int)lds_byte_off);
  const unsigned ve    = (unsigned)__builtin_amdgcn_readfirstlane((int)valid_elems);

  v4u g0;
  g0[0] = 1u;                                      // count=1, user-mode descriptor
  g0[1] = lof;                                     // lds_addr (bytes)
  g0[2] = ga_lo;                                   // global_addr[31:0]
  g0[3] = (ga_hi & 0x01ffffffu) | (2u << 30);      // global_addr[56:32] | type=2

  v8i g1;
  g1[0] = (int)(2u << 16);                         // data_size = 4 bytes
  g1[1] = (int)((ve & 0xffffu) << 16);             // tensor_dim0[15:0]
  g1[2] = (int)((ve >> 16) | (1u << 16));          // tensor_dim0[31:16] | tensor_dim1=1
  g1[3] = (int)((tile_elems & 0xffffu) << 16);     // tile_dim0
  g1[4] = 1;                                       // tile_dim1 = 1 (tile_dim2 = 0)
  g1[5] = (int)ve;                                 // tensor_dim0_stride[31:0]
  g1[6] = 0;                                       // stride hi | tensor_dim1_stride lo
  g1[7] = 0;

  v4i g2 = {0, 0, 0, 0};
  v4i g3 = {0, 0, 0, 0};
  v8i g4 = {0, 0, 0, 0, 0, 0, 0, 0};               // extra group (unused, <=2D)
  __builtin_amdgcn_tensor_load_to_lds(g0, g1, g2, g3, g4, 0);
}

__device__ __forceinline__ unsigned lds_off(const void* p) {
  // Generic pointer to an LDS object: low 32 bits are the LDS byte offset.
  return (unsigned)(unsigned long long)p;
}

__device__ __forceinline__ float resid_sq(float lh, int d, int ev,
                                          const float* __restrict__ base) {
  const float p = base[d] * __expf(lh);
  const float r = p - (float)ev;
  return r * r;
}

// ---------------------------------------------------------------------------
// Kernel 3: MSE pass. Tiles of all three arrays are staged into LDS by the
// Tensor Data Mover (wave 0 issues 3 descriptors per tile, double-buffered;
// s_wait_tensorcnt(3) keeps the next tile's DMA in flight while tile t is
// consumed). Residual squares are summed on the matrix pipe with
// V_WMMA_F32_16X16X4_F32, B = ones, C accumulating across iterations — full
// f32 precision, VALU stays free for v_exp_f32. Uniform trip count +
// zero-padded lanes keep EXEC all-1s at every WMMA.
// ---------------------------------------------------------------------------
__global__ void mse_wmma_kernel(const float* __restrict__ log_h,
                                const int* __restrict__ durations,
                                const int* __restrict__ events,
                                const float* __restrict__ base,
                                float* __restrict__ sse,
                                int n) {
  __shared__ float s_lh[2][TILE];
  __shared__ int   s_d [2][TILE];
  __shared__ int   s_e [2][TILE];

  const int ntiles = (n + TILE - 1) / TILE;
  // Wave-uniform scalar condition -> guaranteed s_cbranch, so only wave 0
  // issues TDM (tensor ops ignore EXEC and would otherwise issue 8x).
  const bool issuer =
      (__builtin_amdgcn_readfirstlane((int)(threadIdx.x >> 5)) == 0);

  v8f acc = {0.0f, 0.0f, 0.0f, 0.0f, 0.0f, 0.0f, 0.0f, 0.0f};
  const v2f ones_b = {1.0f, 1.0f};

  int t = blockIdx.x;
  int parity = 0;

  if (t < ntiles && issuer) {                      // prologue: fetch first tile
    const int start = t * TILE;
    const unsigned rem = (unsigned)(n - start);
    tdm_load_1d(log_h + start,     lds_off(&s_lh[0][0]), rem, TILE);
    tdm_load_1d(durations + start, lds_off(&s_d [0][0]), rem, TILE);
    tdm_load_1d(events + start,    lds_off(&s_e [0][0]), rem, TILE);
  }

  for (; t < ntiles; t += gridDim.x, parity ^= 1) {
    const int tn = t + gridDim.x;
    if (issuer) {
      if (tn < ntiles) {                           // prefetch next tile
        const int start = tn * TILE;
        const unsigned rem = (unsigned)(n - start);
        const int pb = parity ^ 1;
        tdm_load_1d(log_h + start,     lds_off(&s_lh[pb][0]), rem, TILE);
        tdm_load_1d(durations + start, lds_off(&s_d [pb][0]), rem, TILE);
        tdm_load_1d(events + start,    lds_off(&s_e [pb][0]), rem, TILE);
        __builtin_amdgcn_s_wait_tensorcnt(3);      // tile t done, t+1 in flight
      } else {
        __builtin_amdgcn_s_wait_tensorcnt(0);
      }
    }
    __syncthreads();                               // tile t visible to all waves

    const int start = t * TILE;
#pragma unroll
    for (int it = 0; it < TILE / (THREADS * 2); ++it) {
      const int j0 = (int)threadIdx.x + it * (THREADS * 2);
      const int j1 = j0 + THREADS;
      float q0 = 0.0f, q1 = 0.0f;
      if (start + j0 < n)
        q0 = resid_sq(s_lh[parity][j0], s_d[parity][j0], s_e[parity][j0], base);
      if (start + j1 < n)
        q1 = resid_sq(s_lh[parity][j1], s_d[parity][j1], s_e[parity][j1], base);
      v2f a = {q0, q1};
      acc = __builtin_amdgcn_wmma_f32_16x16x4_f32(
          /*neg_a=*/false, a, /*neg_b=*/false, ones_b,
          /*c_mod=*/(short)0, acc, /*reuse_a=*/false, /*reuse_b=*/false);
    }
    __syncthreads();                               // done reading before reuse
  }

  // C/D layout: column N=0 lives in lane 0 (M=0..7) and lane 16 (M=8..15).
  const int lane = threadIdx.x & 31;               // wave32
  if (lane == 0 || lane == 16) {
    const float s = acc[0] + acc[1] + acc[2] + acc[3] +
                    acc[4] + acc[5] + acc[6] + acc[7];
    unsafeAtomicAdd(sse, s);
  }
}

// ---------------------------------------------------------------------------
// Kernel 4: scalar finalize with the no-events gate.
// ---------------------------------------------------------------------------
__global__ void finalize_kernel(const float* __restrict__ sse,
                                const float* __restrict__ tot_ev,
                                float* __restrict__ out,
                                float inv_n) {
  if (blockIdx.x == 0 && threadIdx.x == 0) {
    out[0] = (tot_ev[0] == 0.0f) ? 0.0f : sse[0] * inv_n;
  }
}

// ---------------------------------------------------------------------------
extern "C" void kernel_launch(void* const* d_in, const int* in_sizes, int n_in,
                              void* d_out, int out_size, void* d_ws, size_t ws_size,
                              hipStream_t stream) {
  const float* log_h     = (const float*)d_in[0];
  const int*   durations = (const int*)d_in[1];
  const int*   events    = (const int*)d_in[2];
  float*       out       = (float*)d_out;
  const int    n         = in_sizes[0];

  // Workspace layout (floats): expg_sum[K] | ev_sum[K] | base[K] | sse | tot_ev
  float* g_exp = (float*)d_ws;
  float* g_ev  = g_exp + K_SEG;
  float* base  = g_ev + K_SEG;
  float* sse   = base + K_SEG;
  float* tot   = sse + 1;

  const int zn = 3 * K_SEG + 2;
  zero_ws_kernel<<<(zn + THREADS - 1) / THREADS, THREADS, 0, stream>>>(g_exp, zn);

  seg_hist_kernel<<<BLOCKS, THREADS, 2 * K_SEG * sizeof(float), stream>>>(
      log_h, durations, events, g_exp, g_ev, n);

  baseline_kernel<<<1, 1024, 0, stream>>>(g_exp, g_ev, base, tot);

  mse_wmma_kernel<<<BLOCKS, THREADS, 0, stream>>>(
      log_h, durations, events, base, sse, n);

  finalize_kernel<<<1, 1, 0, stream>>>(sse, tot, out, 1.0f / (float)n);
}